// LiquidNeuron_18769007084256
// MI455X (gfx1250) — compile-verified
//
#include <hip/hip_runtime.h>
#include <hip/hip_bf16.h>
#include <math.h>

#define DIN   1024
#define DOUT  1024
#define BATCH 16384
#define BASE_LR 0.1f
#define LN_EPS  1e-5f

typedef __attribute__((ext_vector_type(2))) float v2f;
typedef __attribute__((ext_vector_type(8))) float v8f;

static __device__ __forceinline__ v8f wmma_f32(v2f a, v2f b, v8f c) {
    return __builtin_amdgcn_wmma_f32_16x16x4_f32(false, a, false, b,
                                                 (short)0, c, false, false);
}

// ---------------------------------------------------------------------------
// Kernel 0: zero scratch (hebb accumulator + msq column sums)
// ---------------------------------------------------------------------------
__global__ void ln_zero_f32(float* __restrict__ p, int n) {
    int i = blockIdx.x * blockDim.x + threadIdx.x;
    if (i < n) p[i] = 0.0f;
}

// ---------------------------------------------------------------------------
// Kernel 1: eff_lr = mean(plasticity) * BASE_LR  (single block)
// ---------------------------------------------------------------------------
__global__ __launch_bounds__(256) void ln_plast_mean(const float* __restrict__ p,
                                                     float* __restrict__ eff_lr) {
    __shared__ float s[256];
    float sum = 0.0f;
    for (int i = threadIdx.x; i < BATCH; i += 256) sum += p[i];
    s[threadIdx.x] = sum;
    __syncthreads();
    for (int ofs = 128; ofs > 0; ofs >>= 1) {
        if ((int)threadIdx.x < ofs) s[threadIdx.x] += s[threadIdx.x + ofs];
        __syncthreads();
    }
    if (threadIdx.x == 0) *eff_lr = s[0] * (BASE_LR / (float)BATCH);
}

// ---------------------------------------------------------------------------
// Kernel 2: dual GEMM  slow = x@Ws^T, fast = x@Wf^T.
// Block: 8 waves share the SAME 16 batch rows (A stream), 512 consecutive
// output columns. Wave tile: 16(batch) x 64(out) for BOTH weight matrices.
// 4-deep software pipeline (256 K-stages divide evenly by 4): each buffer's
// refill is 4 stages away from its consumer, so no compiler copy-chains are
// needed and load latency hides behind 3 stages of WMMAs.
// ---------------------------------------------------------------------------
__global__ __launch_bounds__(256) void ln_gemm_dual(const float* __restrict__ x,
                                                    const float* __restrict__ Ws,
                                                    const float* __restrict__ Wf,
                                                    float* __restrict__ slow,
                                                    float* __restrict__ fast,
                                                    float* __restrict__ msq_sum) {
    const int lane  = threadIdx.x & 31;
    const int wave  = threadIdx.x >> 5;
    const int tileM = blockIdx.x >> 1;                 // 1024 M-tiles
    const int bm    = tileM * 16;
    const int on    = (blockIdx.x & 1) * 512 + wave * 64;
    const int r     = lane & 15;
    const int h     = lane >> 4;

    // A: x[bm + r, k + 2h + {0,1}]  (consecutive pair -> b64 load)
    const float* __restrict__ arow = x + (size_t)(bm + r) * DIN + 2 * h;
    // B (= W^T): W[on + j*16 + r, k + 2h + {0,1}]
    const float* bp[8];
#pragma unroll
    for (int j = 0; j < 4; ++j) {
        bp[j]     = Ws + (size_t)(on + j * 16 + r) * DIN + 2 * h;
        bp[4 + j] = Wf + (size_t)(on + j * 16 + r) * DIN + 2 * h;
    }

    v8f accS[4] = {}, accF[4] = {};

    auto loadS = [&](v2f& A, v2f (&B)[8], int ks) {
        A = *(const v2f*)(arow + ks);
#pragma unroll
        for (int j = 0; j < 8; ++j) B[j] = *(const v2f*)(bp[j] + ks);
    };
    auto compS = [&](const v2f& A, const v2f (&B)[8]) {
#pragma unroll
        for (int j = 0; j < 4; ++j) {
            accS[j] = wmma_f32(A, B[j],     accS[j]);
            accF[j] = wmma_f32(A, B[4 + j], accF[j]);
        }
    };

    v2f A0, A1, A2, A3;
    v2f B0[8], B1[8], B2[8], B3[8];
    loadS(A0, B0, 0);
    loadS(A1, B1, 4);
    loadS(A2, B2, 8);
    loadS(A3, B3, 12);

    for (int k = 0; k < DIN - 16; k += 16) {
        compS(A0, B0); loadS(A0, B0, k + 16);
        compS(A1, B1); loadS(A1, B1, k + 20);
        compS(A2, B2); loadS(A2, B2, k + 24);
        compS(A3, B3); loadS(A3, B3, k + 28);
    }
    compS(A0, B0);
    compS(A1, B1);
    compS(A2, B2);
    compS(A3, B3);

    // D layout: VGPR rr, half h -> M = rr + 8h, N = r
#pragma unroll
    for (int j = 0; j < 4; ++j) {
        const int col = on + j * 16 + r;
        float ss = 0.0f;
#pragma unroll
        for (int rr = 0; rr < 8; ++rr) {
            const size_t row = (size_t)(bm + rr + 8 * h);
            slow[row * DOUT + col] = accS[j][rr];
            const float f = accF[j][rr];
            fast[row * DOUT + col] = f;
            ss += f * f;
        }
        atomicAdd(&msq_sum[col], ss);   // partial sum over 8 batch rows
    }
}

// ---------------------------------------------------------------------------
// Kernel 3: hebb accumulation  H += fast^T @ x  (split-K over 16384 batch).
// A[m=o, k=b] = fast[b, o]; B[k=b, n=i] = x[b, i]. 16 K-splits x (64x16)
// tiles = 16384 waves. Same 4-deep pipeline (strided b32 loads are
// unavoidable: TR-transpose loads only exist for <=16-bit elements).
// ---------------------------------------------------------------------------
__global__ __launch_bounds__(256) void ln_gemm_hebb(const float* __restrict__ fast,
                                                    const float* __restrict__ x,
                                                    float* __restrict__ hebb) {
    const int lane   = threadIdx.x & 31;
    const int wave   = threadIdx.x >> 5;
    const int waveId = blockIdx.x * 8 + wave;
    const int NT     = DIN / 64;                       // 16
    const int MT     = DOUT / 16;                      // 64
    int t            = waveId;
    const int split  = t / (NT * MT);                  // 0..15
    t                = t % (NT * MT);
    const int tileN  = t % NT;
    const int tileM  = t / NT;
    const int om     = tileM * 16;
    const int in     = tileN * 64;
    const int KCH    = BATCH / 16;                     // 1024 rows per split
    const int k0     = split * KCH;
    const int r      = lane & 15;
    const int h      = lane >> 4;

    const float* __restrict__ fcol = fast + (size_t)(k0 + 2 * h) * DOUT + om + r;
    const float* __restrict__ xcol = x + (size_t)(k0 + 2 * h) * DIN + in + r;

    v8f acc[4] = {};

    auto loadH = [&](v2f& A, v2f (&B)[4], int ks) {
        const float* fb = fcol + (size_t)ks * DOUT;
        const float* xb = xcol + (size_t)ks * DIN;
        A.x = fb[0];
        A.y = fb[DOUT];
#pragma unroll
        for (int j = 0; j < 4; ++j) {
            B[j].x = xb[j * 16];
            B[j].y = xb[DIN + j * 16];
        }
    };
    auto compH = [&](const v2f& A, const v2f (&B)[4]) {
#pragma unroll
        for (int j = 0; j < 4; ++j) acc[j] = wmma_f32(A, B[j], acc[j]);
    };

    v2f A0, A1, A2, A3;
    v2f B0[4], B1[4], B2[4], B3[4];
    loadH(A0, B0, 0);
    loadH(A1, B1, 4);
    loadH(A2, B2, 8);
    loadH(A3, B3, 12);

    for (int k = 0; k < KCH - 16; k += 16) {
        compH(A0, B0); loadH(A0, B0, k + 16);
        compH(A1, B1); loadH(A1, B1, k + 20);
        compH(A2, B2); loadH(A2, B2, k + 24);
        compH(A3, B3); loadH(A3, B3, k + 28);
    }
    compH(A0, B0);
    compH(A1, B1);
    compH(A2, B2);
    compH(A3, B3);

#pragma unroll
    for (int j = 0; j < 4; ++j)
#pragma unroll
        for (int rr = 0; rr < 8; ++rr)
            atomicAdd(&hebb[(size_t)(om + rr + 8 * h) * DIN + in + j * 16 + r],
                      acc[j][rr]);
}

// ---------------------------------------------------------------------------
// Kernel 4: W_fast_new = Wf + tanh(hebb/B - (msq/B)*Wf) * eff_lr
// ---------------------------------------------------------------------------
__global__ __launch_bounds__(256) void ln_wfast_update(const float* __restrict__ Wf,
                                                       const float* __restrict__ hebb_sum,
                                                       const float* __restrict__ msq_sum,
                                                       const float* __restrict__ eff_lr,
                                                       float* __restrict__ Wnew) {
    const int i = blockIdx.x * 256 + threadIdx.x;      // 0 .. 1M-1
    const int o = i / DIN;
    const float w      = Wf[i];
    const float hebb   = hebb_sum[i] * (1.0f / (float)BATCH);
    const float forget = (msq_sum[o] * (1.0f / (float)BATCH)) * w;
    Wnew[i] = w + tanhf(hebb - forget) * (*eff_lr);
}

// ---------------------------------------------------------------------------
// Kernel 5: LayerNorm. One wave per row (32 elems/lane in registers, wave32
// shfl_xor reductions). Reads slow + fast (fast lives in the out region),
// overwrites the out region in place.
// ---------------------------------------------------------------------------
__global__ __launch_bounds__(256) void ln_layernorm(const float* __restrict__ slow,
                                                    float* __restrict__ outfast,
                                                    const float* __restrict__ alpha,
                                                    const float* __restrict__ gamma,
                                                    const float* __restrict__ beta) {
    const int lane = threadIdx.x & 31;
    const int wave = threadIdx.x >> 5;
    const int row  = blockIdx.x * 8 + wave;
    const float a  = alpha[0];
    const float* __restrict__ srow = slow + (size_t)row * DOUT;
    float* __restrict__ frow       = outfast + (size_t)row * DOUT;

    float c[32];
    float sum = 0.0f;
#pragma unroll
    for (int t = 0; t < 32; ++t) {
        const int idx = lane + 32 * t;
        const float v = srow[idx] + frow[idx] * a;
        c[t] = v;
        sum += v;
    }
#pragma unroll
    for (int o = 16; o > 0; o >>= 1) sum += __shfl_xor(sum, o, 32);
    const float mu = sum * (1.0f / (float)DOUT);

    float vs = 0.0f;
#pragma unroll
    for (int t = 0; t < 32; ++t) { const float d = c[t] - mu; vs += d * d; }
#pragma unroll
    for (int o = 16; o > 0; o >>= 1) vs += __shfl_xor(vs, o, 32);
    const float rstd = rsqrtf(vs * (1.0f / (float)DOUT) + LN_EPS);

#pragma unroll
    for (int t = 0; t < 32; ++t) {
        const int idx = lane + 32 * t;
        frow[idx] = (c[t] - mu) * rstd * gamma[idx] + beta[idx];
    }
}

// ---------------------------------------------------------------------------
extern "C" void kernel_launch(void* const* d_in, const int* in_sizes, int n_in,
                              void* d_out, int out_size, void* d_ws, size_t ws_size,
                              hipStream_t stream) {
    const float* x     = (const float*)d_in[0];
    const float* plast = (const float*)d_in[1];
    const float* alpha = (const float*)d_in[2];
    const float* Ws    = (const float*)d_in[3];
    const float* Wf    = (const float*)d_in[4];
    const float* gamma = (const float*)d_in[5];
    const float* beta  = (const float*)d_in[6];

    // d_out layout: [out (B*DOUT)] [slow (B*DOUT)] [W_fast_new (DOUT*DIN)]
    float* out  = (float*)d_out;                       // holds `fast` until LN
    float* slow = out + (size_t)BATCH * DOUT;
    float* wnew = slow + (size_t)BATCH * DOUT;

    // workspace: hebb accumulator (1M), msq column sums (1024), eff_lr (1)
    float* hebb  = (float*)d_ws;
    float* msq   = hebb + (size_t)DOUT * DIN;
    float* efflr = msq + DOUT;

    const int nz = DOUT * DIN + DOUT;                  // hebb + msq contiguous
    ln_zero_f32<<<(nz + 255) / 256, 256, 0, stream>>>(hebb, nz);
    ln_plast_mean<<<1, 256, 0, stream>>>(plast, efflr);

    // 1024 M-tiles x 2 column halves = 2048 blocks (8 waves/block)
    ln_gemm_dual<<<(BATCH / 16) * 2, 256, 0, stream>>>(x, Ws, Wf, slow, out, msq);

    // 16 K-splits * 64 * 16 tiles = 16384 wave tiles -> 2048 blocks
    ln_gemm_hebb<<<16 * (DOUT / 16) * (DIN / 64) / 8, 256, 0, stream>>>(
        out, x, hebb);

    ln_wfast_update<<<(DOUT * DIN) / 256, 256, 0, stream>>>(
        Wf, hebb, msq, efflr, wnew);

    ln_layernorm<<<BATCH / 8, 256, 0, stream>>>(slow, out, alpha, gamma, beta);
}